// kmeans_distribution_layer_58162447123065
// MI455X (gfx1250) — compile-verified
//
#include <hip/hip_runtime.h>
#include <hip/hip_bf16.h>

typedef __attribute__((ext_vector_type(2))) float v2f;
typedef __attribute__((ext_vector_type(8))) float v8f;
typedef __attribute__((ext_vector_type(4))) unsigned u32x4;
typedef __attribute__((ext_vector_type(8))) int i32x8;
typedef __attribute__((ext_vector_type(4))) int i32x4;

#define B_N 2048
#define K_N 64
#define D_N 512

#if __has_builtin(__builtin_amdgcn_tensor_load_to_lds) && __has_builtin(__builtin_amdgcn_s_wait_tensorcnt)
#define HAVE_TDM 1
#else
#define HAVE_TDM 0
#endif

// workspace layout (float offsets)
#define OFF_DOT    0         // [B_N*K_N]
#define OFF_DIST   131072    // [B_N*K_N]
#define OFF_XX     262144    // [B_N]
#define OFF_CC     264192    // [K_N]
#define OFF_ASSIGN 264256    // [B_N] (int)
#define OFF_CNT    266304    // [K_N]
#define OFF_MTOT   266368    // [K_N*D_N]
#define OFF_MEAN   299136    // [K_N*D_N]
#define OFF_SUMSQ  331904    // [K_N*D_N]
#define OFF_DOTMU  364672    // [B_N]
#define OFF_MUSQ   366720    // [K_N]
#define OFF_WEI    366784    // [K_N]
#define OFF_FRO    366848    // [K_N]
#define WS_END     366912
#define ZERO_BEG   OFF_CNT
#define ZERO_N     (WS_END - OFF_CNT)

#if HAVE_TDM
// low 32 bits of a generic LDS address are the wave-relative LDS byte offset
__device__ __forceinline__ unsigned lds_off_of(const void* p) {
  return (unsigned)(unsigned long long)p;
}

// 2D TDM load: tile_d1 rows x tile_d0 elems (f32) from a row-major tensor with
// row stride `stride0` elems, landing compact row-major at lds_off.
__device__ __forceinline__ void tdm_load_2d(unsigned lds_off, const float* gptr,
                                            unsigned tensor_d0, unsigned tensor_d1,
                                            unsigned tile_d0, unsigned tile_d1,
                                            unsigned stride0) {
  unsigned long long ga = (unsigned long long)gptr;  // byte address
  u32x4 g0;
  g0[0] = 1u;                                        // count=1, is_restore=0, gather off
  g0[1] = lds_off;                                   // lds_addr
  g0[2] = (unsigned)ga;                              // global_addr[31:0]
  g0[3] = (unsigned)((ga >> 32) & 0x01FFFFFFu) | (2u << 30);  // addr[56:32] | type=2
  i32x8 g1;
  g1[0] = (int)(2u << 16);                           // workgroup_mask=0, data_size=2 (4B)
  g1[1] = (int)((tensor_d0 & 0xFFFFu) << 16);        // tensor_dim0[15:0] -> bits[63:48]
  g1[2] = (int)(((tensor_d0 >> 16) & 0xFFFFu) | ((tensor_d1 & 0xFFFFu) << 16));
  g1[3] = (int)(((tensor_d1 >> 16) & 0xFFFFu) | ((tile_d0 & 0xFFFFu) << 16));
  g1[4] = (int)(tile_d1 & 0xFFFFu);                  // tile_dim1; tile_dim2=0
  g1[5] = (int)stride0;                              // tensor_dim0_stride[31:0]
  g1[6] = 0;                                         // stride0 hi / stride1 lo
  g1[7] = 0;
  i32x4 z4 = {0, 0, 0, 0};
  i32x8 z8 = {0, 0, 0, 0, 0, 0, 0, 0};
  __builtin_amdgcn_tensor_load_to_lds(g0, g1, z4, z4, z8, 0);
}
#endif

// ---------------------------------------------------------------- zero scratch
__global__ void kzero(float* __restrict__ p, int n) {
  int i = blockIdx.x * blockDim.x + threadIdx.x;
  if (i < n) p[i] = 0.0f;
}

// ---------------------------------------------------------------- row sqnorms
__global__ __launch_bounds__(128) void ksqnorm(const float* __restrict__ src,
                                               float* __restrict__ dst) {
  __shared__ float s[128];
  int row = blockIdx.x;
  int t = threadIdx.x;
  float a = 0.0f;
  for (int d = t; d < D_N; d += 128) {
    float v = src[row * D_N + d];
    a += v * v;
  }
  s[t] = a;
  __syncthreads();
  for (int w = 64; w > 0; w >>= 1) {
    if (t < w) s[t] += s[t + w];
    __syncthreads();
  }
  if (t == 0) dst[row] = s[0];
}

// ---------------------------------------------------------------- dot = x @ centers^T  (WMMA f32 16x16x4)
// block: 128 threads (4 waves). Block covers 16 rows of x vs all 64 centers.
// TDM double-buffers the LDS tiles; wave 0 drives the DMA.
__global__ __launch_bounds__(128) void kgemm_xct(const float* __restrict__ x,
                                                 const float* __restrict__ c,
                                                 float* __restrict__ dot) {
  __shared__ float xs[2][16 * 64];   // [buf][row][kk]
  __shared__ float cs[2][64 * 64];   // [buf][n][kk]
  const int bm = blockIdx.x * 16;
  const int tid = threadIdx.x;
  const int wave = tid >> 5;
  const int lane = tid & 31;
  const int half = lane >> 4;
  const int l16 = lane & 15;

  v8f acc = {0.f, 0.f, 0.f, 0.f, 0.f, 0.f, 0.f, 0.f};

#if HAVE_TDM
  if (wave == 0) {
    tdm_load_2d(lds_off_of(&xs[0][0]), x + (size_t)bm * D_N, D_N, B_N, 64, 16, D_N);
    tdm_load_2d(lds_off_of(&cs[0][0]), c, D_N, K_N, 64, 64, D_N);
    __builtin_amdgcn_s_wait_tensorcnt(0);
  }
  __syncthreads();
  for (int cidx = 0; cidx < D_N / 64; ++cidx) {
    const int cur = cidx & 1;
    if (wave == 0 && cidx + 1 < D_N / 64) {   // prefetch next chunk via TDM
      int ch = (cidx + 1) * 64;
      tdm_load_2d(lds_off_of(&xs[cur ^ 1][0]), x + (size_t)bm * D_N + ch, D_N, B_N, 64, 16, D_N);
      tdm_load_2d(lds_off_of(&cs[cur ^ 1][0]), c + ch, D_N, K_N, 64, 64, D_N);
    }
    const float* xsb = xs[cur];
    const float* csb = cs[cur];
#pragma unroll
    for (int kk = 0; kk < 64; kk += 4) {
      v2f a;
      a.x = xsb[l16 * 64 + kk + 2 * half + 0];
      a.y = xsb[l16 * 64 + kk + 2 * half + 1];
      v2f b;
      b.x = csb[(wave * 16 + l16) * 64 + kk + 2 * half + 0];
      b.y = csb[(wave * 16 + l16) * 64 + kk + 2 * half + 1];
      acc = __builtin_amdgcn_wmma_f32_16x16x4_f32(false, a, false, b,
                                                  (short)0, acc, false, false);
    }
    __syncthreads();
    if (wave == 0 && cidx + 1 < D_N / 64) __builtin_amdgcn_s_wait_tensorcnt(0);
    __syncthreads();
  }
#else
  for (int ch = 0; ch < D_N; ch += 64) {
    for (int i = tid; i < 16 * 64; i += 128) {
      int r = i >> 6, q = i & 63;
      xs[0][i] = x[(bm + r) * D_N + ch + q];
    }
    for (int i = tid; i < 64 * 64; i += 128) {
      int r = i >> 6, q = i & 63;
      cs[0][i] = c[r * D_N + ch + q];
    }
    __syncthreads();
#pragma unroll
    for (int kk = 0; kk < 64; kk += 4) {
      v2f a;
      a.x = xs[0][l16 * 64 + kk + 2 * half + 0];
      a.y = xs[0][l16 * 64 + kk + 2 * half + 1];
      v2f b;
      b.x = cs[0][(wave * 16 + l16) * 64 + kk + 2 * half + 0];
      b.y = cs[0][(wave * 16 + l16) * 64 + kk + 2 * half + 1];
      acc = __builtin_amdgcn_wmma_f32_16x16x4_f32(false, a, false, b,
                                                  (short)0, acc, false, false);
    }
    __syncthreads();
  }
#endif
  // C layout: VGPR r -> M = r + 8*half, N = l16
#pragma unroll
  for (int r = 0; r < 8; ++r) {
    int m = r + 8 * half;
    dot[(bm + m) * K_N + wave * 16 + l16] = acc[r];
  }
}

// ---------------------------------------------------------------- per-row: dist, softmaxes, argmin
__global__ __launch_bounds__(64) void krow(const float* __restrict__ dot,
                                           const float* __restrict__ xx,
                                           const float* __restrict__ cc,
                                           float* __restrict__ dist,
                                           int* __restrict__ assign,
                                           float* __restrict__ cnt,
                                           float* __restrict__ out_rk,
                                           float* __restrict__ out_rc,
                                           float* __restrict__ out_cd) {
  __shared__ float sm[64];
  __shared__ int si[64];
  const int b = blockIdx.x;
  const int k = threadIdx.x;

  float d = xx[b] + cc[k] - 2.0f * dot[b * K_N + k];
  dist[b * K_N + k] = d;

  float e1 = -0.5f * d;
  sm[k] = e1; __syncthreads();
  for (int w = 32; w > 0; w >>= 1) { if (k < w) sm[k] = fmaxf(sm[k], sm[k + w]); __syncthreads(); }
  float m1 = sm[0]; __syncthreads();
  float n1 = __expf(e1 - m1);
  sm[k] = n1; __syncthreads();
  for (int w = 32; w > 0; w >>= 1) { if (k < w) sm[k] += sm[k + w]; __syncthreads(); }
  float s1 = sm[0]; __syncthreads();
  out_rk[b * K_N + k] = __logf(fmaxf(n1 / s1, 1e-8f));

  float e2 = __powf(d + 1.0f, -0.5f * (1.0f + (float)D_N));
  sm[k] = e2; __syncthreads();
  for (int w = 32; w > 0; w >>= 1) { if (k < w) sm[k] = fmaxf(sm[k], sm[k + w]); __syncthreads(); }
  float m2 = sm[0]; __syncthreads();
  float n2 = __expf(e2 - m2);
  sm[k] = n2; __syncthreads();
  for (int w = 32; w > 0; w >>= 1) { if (k < w) sm[k] += sm[k + w]; __syncthreads(); }
  float s2 = sm[0]; __syncthreads();
  out_rc[b * K_N + k] = __logf(fmaxf(n2 / s2, 1e-8f));

  sm[k] = d; si[k] = k; __syncthreads();
  for (int w = 32; w > 0; w >>= 1) {
    if (k < w) {
      if (sm[k + w] < sm[k]) { sm[k] = sm[k + w]; si[k] = si[k + w]; }
      else if (sm[k + w] == sm[k] && si[k + w] < si[k]) { si[k] = si[k + w]; }
    }
    __syncthreads();
  }
  if (k == 0) {
    assign[b] = si[0];
    out_cd[b] = sqrtf(fmaxf(sm[0], 0.0f));
    atomicAdd(&cnt[si[0]], 1.0f);
  }
}

// ---------------------------------------------------------------- scatter x into per-cluster totals
__global__ __launch_bounds__(128) void kscatter_mean(const float* __restrict__ x,
                                                     const int* __restrict__ assign,
                                                     float* __restrict__ mtot) {
  const int b = blockIdx.x;
  const int a = assign[b];
  for (int d = threadIdx.x; d < D_N; d += 128)
    atomicAdd(&mtot[a * D_N + d], x[b * D_N + d]);
}

// ---------------------------------------------------------------- means, ||mu||^2, weighted center dist
__global__ __launch_bounds__(256) void kfinish_mean(const float* __restrict__ mtot,
                                                    const float* __restrict__ cnt,
                                                    const float* __restrict__ centers,
                                                    float* __restrict__ mean,
                                                    float* __restrict__ musq,
                                                    float* __restrict__ wei) {
  __shared__ float s1[256];
  __shared__ float s2[256];
  const int k = blockIdx.x;
  const int t = threadIdx.x;
  const float ce = cnt[k] + 1e-7f;
  float a1 = 0.f, a2 = 0.f;
  for (int d = t; d < D_N; d += 256) {
    float m = mtot[k * D_N + d] / ce;
    mean[k * D_N + d] = m;
    a1 += m * m;
    float df = m - centers[k * D_N + d];
    a2 += df * df;
  }
  s1[t] = a1; s2[t] = a2; __syncthreads();
  for (int w = 128; w > 0; w >>= 1) {
    if (t < w) { s1[t] += s1[t + w]; s2[t] += s2[t + w]; }
    __syncthreads();
  }
  if (t == 0) { musq[k] = s1[0]; wei[k] = cnt[k] * s2[0]; }
}

// ---------------------------------------------------------------- diag sums + mu.x per row
__global__ __launch_bounds__(256) void kcenter_stats(const float* __restrict__ x,
                                                     const int* __restrict__ assign,
                                                     const float* __restrict__ mean,
                                                     float* __restrict__ sumsq,
                                                     float* __restrict__ dotmu) {
  __shared__ float s[256];
  const int b = blockIdx.x;
  const int a = assign[b];
  const int t = threadIdx.x;
  float acc = 0.f;
  for (int d = t; d < D_N; d += 256) {
    float xv = x[b * D_N + d];
    float mv = mean[a * D_N + d];
    float v = xv - mv;
    atomicAdd(&sumsq[a * D_N + d], v * v);
    acc += mv * xv;
  }
  s[t] = acc; __syncthreads();
  for (int w = 128; w > 0; w >>= 1) {
    if (t < w) s[t] += s[t + w];
    __syncthreads();
  }
  if (t == 0) dotmu[b] = s[0];
}

// ---------------------------------------------------------------- Gram tile (WMMA) -> Frobenius per cluster
// block: 128 threads (4 waves). Covers 16 rows (bi) x 64 cols (bj) of G = X X^T.
__global__ __launch_bounds__(128) void kgram_fro(const float* __restrict__ x,
                                                 const int* __restrict__ assign,
                                                 const float* __restrict__ dotmu,
                                                 const float* __restrict__ musq,
                                                 float* __restrict__ fro) {
  __shared__ float xi[2][16 * 64];
  __shared__ float xj[2][64 * 64];
  __shared__ float frosh[64];
  const int bi0 = blockIdx.x * 16;
  const int bj0 = blockIdx.y * 64;
  const int tid = threadIdx.x;
  const int wave = tid >> 5;
  const int lane = tid & 31;
  const int half = lane >> 4;
  const int l16 = lane & 15;

  if (tid < 64) frosh[tid] = 0.0f;

  v8f acc = {0.f, 0.f, 0.f, 0.f, 0.f, 0.f, 0.f, 0.f};

#if HAVE_TDM
  if (wave == 0) {
    tdm_load_2d(lds_off_of(&xi[0][0]), x + (size_t)bi0 * D_N, D_N, B_N, 64, 16, D_N);
    tdm_load_2d(lds_off_of(&xj[0][0]), x + (size_t)bj0 * D_N, D_N, B_N, 64, 64, D_N);
    __builtin_amdgcn_s_wait_tensorcnt(0);
  }
  __syncthreads();
  for (int cidx = 0; cidx < D_N / 64; ++cidx) {
    const int cur = cidx & 1;
    if (wave == 0 && cidx + 1 < D_N / 64) {
      int ch = (cidx + 1) * 64;
      tdm_load_2d(lds_off_of(&xi[cur ^ 1][0]), x + (size_t)bi0 * D_N + ch, D_N, B_N, 64, 16, D_N);
      tdm_load_2d(lds_off_of(&xj[cur ^ 1][0]), x + (size_t)bj0 * D_N + ch, D_N, B_N, 64, 64, D_N);
    }
    const float* xib = xi[cur];
    const float* xjb = xj[cur];
#pragma unroll
    for (int kk = 0; kk < 64; kk += 4) {
      v2f a;
      a.x = xib[l16 * 64 + kk + 2 * half + 0];
      a.y = xib[l16 * 64 + kk + 2 * half + 1];
      v2f b;
      b.x = xjb[(wave * 16 + l16) * 64 + kk + 2 * half + 0];
      b.y = xjb[(wave * 16 + l16) * 64 + kk + 2 * half + 1];
      acc = __builtin_amdgcn_wmma_f32_16x16x4_f32(false, a, false, b,
                                                  (short)0, acc, false, false);
    }
    __syncthreads();
    if (wave == 0 && cidx + 1 < D_N / 64) __builtin_amdgcn_s_wait_tensorcnt(0);
    __syncthreads();
  }
#else
  for (int ch = 0; ch < D_N; ch += 64) {
    for (int i = tid; i < 16 * 64; i += 128) {
      int r = i >> 6, q = i & 63;
      xi[0][i] = x[(bi0 + r) * D_N + ch + q];
    }
    for (int i = tid; i < 64 * 64; i += 128) {
      int r = i >> 6, q = i & 63;
      xj[0][i] = x[(bj0 + r) * D_N + ch + q];
    }
    __syncthreads();
#pragma unroll
    for (int kk = 0; kk < 64; kk += 4) {
      v2f a;
      a.x = xi[0][l16 * 64 + kk + 2 * half + 0];
      a.y = xi[0][l16 * 64 + kk + 2 * half + 1];
      v2f b;
      b.x = xj[0][(wave * 16 + l16) * 64 + kk + 2 * half + 0];
      b.y = xj[0][(wave * 16 + l16) * 64 + kk + 2 * half + 1];
      acc = __builtin_amdgcn_wmma_f32_16x16x4_f32(false, a, false, b,
                                                  (short)0, acc, false, false);
    }
    __syncthreads();
  }
#endif

  const int bj = bj0 + wave * 16 + l16;
  const int aj = assign[bj];
  const float dmj = dotmu[bj];
#pragma unroll
  for (int r = 0; r < 8; ++r) {
    int bi = bi0 + r + 8 * half;
    int ai = assign[bi];
    if (ai == aj) {
      float t = acc[r] - dotmu[bi] - dmj + musq[ai];
      atomicAdd(&frosh[ai], t * t);   // ds_add_f32
    }
  }
  __syncthreads();
  if (tid < 64) atomicAdd(&fro[tid], frosh[tid]);
}

// ---------------------------------------------------------------- final scalars
__global__ __launch_bounds__(64) void kfinalize(const float* __restrict__ sumsq,
                                                const float* __restrict__ cnt,
                                                const float* __restrict__ fro,
                                                const float* __restrict__ wei,
                                                float* __restrict__ out2) {
  __shared__ float s1[64];
  __shared__ float s2[64];
  const int k = threadIdx.x;
  const float ce = cnt[k] + 1e-7f;
  float diag = 0.f, diagsq = 0.f;
  for (int d = 0; d < D_N; ++d) {
    float cii = sumsq[k * D_N + d] / ce;
    float t = cii - 1.0f;
    diag += t * t;
    diagsq += cii * cii;
  }
  float off = fro[k] / (ce * ce) - diagsq;
  const float BD = (float)B_N * (float)D_N;
  float contrib = cnt[k] * diag / BD + cnt[k] * off / (BD * ((float)D_N - 1.0f));
  s1[k] = wei[k]; s2[k] = contrib; __syncthreads();
  for (int w = 32; w > 0; w >>= 1) {
    if (k < w) { s1[k] += s1[k + w]; s2[k] += s2[k + w]; }
    __syncthreads();
  }
  if (k == 0) {
    out2[0] = s1[0] / ((float)B_N * (float)D_N);  // empirical_mean_mse
    out2[1] = s2[0];                              // empirical_covar_mse
  }
}

// ================================================================ launcher
extern "C" void kernel_launch(void* const* d_in, const int* in_sizes, int n_in,
                              void* d_out, int out_size, void* d_ws, size_t ws_size,
                              hipStream_t stream) {
  (void)in_sizes; (void)n_in; (void)out_size; (void)ws_size;
  const float* x = (const float*)d_in[0];        // [B_N, D_N]
  const float* centers = (const float*)d_in[1];  // [K_N, D_N]
  float* out = (float*)d_out;
  float* ws = (float*)d_ws;

  float* dot    = ws + OFF_DOT;
  float* dist   = ws + OFF_DIST;
  float* xx     = ws + OFF_XX;
  float* cc     = ws + OFF_CC;
  int*   assign = (int*)(ws + OFF_ASSIGN);
  float* cnt    = ws + OFF_CNT;
  float* mtot   = ws + OFF_MTOT;
  float* mean   = ws + OFF_MEAN;
  float* sumsq  = ws + OFF_SUMSQ;
  float* dotmu  = ws + OFF_DOTMU;
  float* musq   = ws + OFF_MUSQ;
  float* wei    = ws + OFF_WEI;
  float* fro    = ws + OFF_FRO;

  float* out_rk = out;                       // resp_kmeans [B_N*K_N]
  float* out_rc = out + B_N * K_N;           // resp_cmm    [B_N*K_N]
  float* out_cd = out + 2 * B_N * K_N;       // cluster_dist [B_N]
  float* out_sc = out + 2 * B_N * K_N + B_N; // two scalars

  kzero<<<(ZERO_N + 255) / 256, 256, 0, stream>>>(ws + ZERO_BEG, ZERO_N);
  ksqnorm<<<B_N, 128, 0, stream>>>(x, xx);
  ksqnorm<<<K_N, 128, 0, stream>>>(centers, cc);
  kgemm_xct<<<B_N / 16, 128, 0, stream>>>(x, centers, dot);
  krow<<<B_N, 64, 0, stream>>>(dot, xx, cc, dist, assign, cnt, out_rk, out_rc, out_cd);
  kscatter_mean<<<B_N, 128, 0, stream>>>(x, assign, mtot);
  kfinish_mean<<<K_N, 256, 0, stream>>>(mtot, cnt, centers, mean, musq, wei);
  kcenter_stats<<<B_N, 256, 0, stream>>>(x, assign, mean, sumsq, dotmu);
  kgram_fro<<<dim3(B_N / 16, B_N / 64), 128, 0, stream>>>(x, assign, dotmu, musq, fro);
  kfinalize<<<1, 64, 0, stream>>>(sumsq, cnt, fro, wei, out_sc);
}